// RegionalConv2Layer_34651796144298
// MI455X (gfx1250) — compile-verified
//
#include <hip/hip_runtime.h>
#include <math.h>

typedef __attribute__((ext_vector_type(2))) float v2f;
typedef __attribute__((ext_vector_type(8))) float v8f;

#define BN 2048
#define TN 2048
#define CN 8
#define AN 27
#define HN 8

// ---------------------------------------------------------------------------
// Stage 1: per-sample hypernetwork MLPs -> kernel[B,8], bias[B]
// One thread per batch row; total work is negligible (<2 MFLOP).
// ---------------------------------------------------------------------------
__global__ __launch_bounds__(256) void mlp_kernel(
    const float* __restrict__ attrs,
    const float* __restrict__ w1,  const float* __restrict__ b1,
    const float* __restrict__ w2,  const float* __restrict__ b2,
    const float* __restrict__ bw1, const float* __restrict__ bb1,
    const float* __restrict__ bw2, const float* __restrict__ bb2,
    const float* __restrict__ mask_attrs,
    const float* __restrict__ mask_hidden,
    float* __restrict__ kern, float* __restrict__ bias)
{
    const int b = blockIdx.x * blockDim.x + threadIdx.x;
    if (b >= BN) return;

    float a[AN];
#pragma unroll
    for (int j = 0; j < AN; ++j)
        a[j] = attrs[b * AN + j] * mask_attrs[b * AN + j];

    float hw[HN], hb[HN];
#pragma unroll
    for (int h = 0; h < HN; ++h) {
        float s1 = b1[h];
        float s2 = bb1[h];
#pragma unroll
        for (int j = 0; j < AN; ++j) {
            s1 = fmaf(a[j], w1[j * HN + h],  s1);
            s2 = fmaf(a[j], bw1[j * HN + h], s2);
        }
        const float mh = mask_hidden[b * HN + h];
        hw[h] = fmaxf(s1, 0.0f) * mh;
        hb[h] = fmaxf(s2, 0.0f) * mh;
    }

#pragma unroll
    for (int c = 0; c < CN; ++c) {
        float s = b2[c];
#pragma unroll
        for (int h = 0; h < HN; ++h)
            s = fmaf(hw[h], w2[h * CN + c], s);
        kern[b * CN + c] = tanhf(s);
    }

    float s = bb2[0];
#pragma unroll
    for (int h = 0; h < HN; ++h)
        s = fmaf(hb[h], bw2[h], s);
    bias[b] = tanhf(s);
}

// ---------------------------------------------------------------------------
// Stage 2: y[b,t] = elu( sum_c x[b,t,c]*kern[b,c] + bias[b] )
// One block per batch row b; 8 waves; each wave eats 16 timesteps per step.
// Channel contraction runs on the matrix pipe: V_WMMA_F32_16X16X4_F32 with a
// broadcast-B tile (kernel[b,k] replicated across the 16 N columns).
//
// A-tile (16x4 f32, ISA 7.12.2): lane m(0..15) holds {x[t0+m, 2*kh+0],
// x[t0+m, 2*kh+1]} where kh = lane>>4. Two WMMAs cover c=0..3 and c=4..7.
// D layout: lane = column N, VGPR r = row M (lanes 0-15 -> M=r, 16-31 -> M=8+r);
// all columns are identical by construction, so lanes 0-7 / 16-23 store rows.
// ---------------------------------------------------------------------------
__global__ __launch_bounds__(256) void conv_elu_kernel(
    const float* __restrict__ x,
    const float* __restrict__ kern,
    const float* __restrict__ bias,
    float* __restrict__ out)
{
    const int b    = blockIdx.x;
    const int tid  = threadIdx.x;
    const int wave = tid >> 5;
    const int lane = tid & 31;
    const int m    = lane & 15;    // row inside the 16-timestep tile
    const int kh   = lane >> 4;    // which K-half this lane supplies
    const int j    = lane & 7;     // accumulator VGPR index for writeout

    const float* __restrict__ xb = x + (size_t)b * (TN * CN);
    const float* __restrict__ kr = kern + b * CN;

    // Broadcast-B tiles: B[k][n] = kern[b,k] for all n.
    // B 4x16 layout mirrors A: VGPR0 = rows K=0 (lanes 0-15) / K=2 (16-31),
    // VGPR1 = rows K=1 / K=3.
    v2f B0, B1;
    B0.x = kh ? kr[2] : kr[0];
    B0.y = kh ? kr[3] : kr[1];
    B1.x = kh ? kr[6] : kr[4];
    B1.y = kh ? kr[7] : kr[5];

    const float bb      = bias[b];
    const bool  writer  = ((lane & 15) < 8);
    const int   m_out   = kh * 8 + j;   // output row this lane emits

    float* __restrict__ ob = out + (size_t)b * TN;

#pragma unroll 2
    for (int tile = wave; tile < TN / 16; tile += 8) {
        const int t0 = tile * 16;
        // 8-byte coalesced loads: wave covers 512 contiguous bytes of x.
        const v2f* p = (const v2f*)(xb + (size_t)(t0 + m) * CN + kh * 2);
        const v2f a0 = p[0];   // channels {2kh, 2kh+1}
        const v2f a1 = p[2];   // channels {4+2kh, 4+2kh+1}

        v8f acc = {};
        acc = __builtin_amdgcn_wmma_f32_16x16x4_f32(
                  false, a0, false, B0, (short)0, acc, false, false);
        acc = __builtin_amdgcn_wmma_f32_16x16x4_f32(
                  false, a1, false, B1, (short)0, acc, false, false);

        if (writer) {
            float v = acc[0];
            v = (j == 1) ? acc[1] : v;
            v = (j == 2) ? acc[2] : v;
            v = (j == 3) ? acc[3] : v;
            v = (j == 4) ? acc[4] : v;
            v = (j == 5) ? acc[5] : v;
            v = (j == 6) ? acc[6] : v;
            v = (j == 7) ? acc[7] : v;
            const float y = v + bb;
            ob[t0 + m_out] = (y > 0.0f) ? y : expm1f(y);
        }
    }
}

// ---------------------------------------------------------------------------
extern "C" void kernel_launch(void* const* d_in, const int* in_sizes, int n_in,
                              void* d_out, int out_size, void* d_ws, size_t ws_size,
                              hipStream_t stream)
{
    const float* x          = (const float*)d_in[0];
    const float* attrs      = (const float*)d_in[1];
    const float* w1         = (const float*)d_in[2];
    const float* b1         = (const float*)d_in[3];
    const float* w2         = (const float*)d_in[4];
    const float* b2         = (const float*)d_in[5];
    const float* bw1        = (const float*)d_in[6];
    const float* bb1        = (const float*)d_in[7];
    const float* bw2        = (const float*)d_in[8];
    const float* bb2        = (const float*)d_in[9];
    const float* mask_attrs = (const float*)d_in[10];
    const float* mask_hid   = (const float*)d_in[11];

    float* kern = (float*)d_ws;          // [B, C]  (64 KB)
    float* bias = kern + (size_t)BN * CN; // [B]     (8 KB)

    mlp_kernel<<<(BN + 255) / 256, 256, 0, stream>>>(
        attrs, w1, b1, w2, b2, bw1, bb1, bw2, bb2,
        mask_attrs, mask_hid, kern, bias);

    conv_elu_kernel<<<BN, 256, 0, stream>>>(
        x, kern, bias, (float*)d_out);
}